// DAGCN_7705171329025
// MI455X (gfx1250) — compile-verified
//
#include <hip/hip_runtime.h>
#include <cmath>

// ---------------- problem constants (match reference) ----------------
#define N_NODES 100000
#define E_EDGES 800000
#define G_GRAPHS 1000
#define D_DIM 128
#define F_INF 64
#define K_HOPS 3
#define NUM_BLOCKS 3
#define H_HEADS 5
#define BN_EPS 1e-5f
#define BN_ROWS 512
#define K_CHUNK 64
#define GEMM_ROW_BLOCKS (N_NODES / 32)   // 3125

typedef float v2f __attribute__((ext_vector_type(2)));
typedef float v8f __attribute__((ext_vector_type(8)));
typedef int v4i_vs __attribute__((vector_size(4 * sizeof(int))));   // matches builtin param
typedef __attribute__((address_space(1))) v4i_vs glb_v4i;
typedef __attribute__((address_space(3))) v4i_vs lds_v4i;

#if defined(__gfx1250__) && __has_builtin(__builtin_amdgcn_global_load_async_to_lds_b128)
#define USE_ASYNC_LDS 1
#else
#define USE_ASYNC_LDS 0
#endif

__device__ __forceinline__ void wait_async_lds() {
#if USE_ASYNC_LDS
#if __has_builtin(__builtin_amdgcn_s_wait_asynccnt)
  __builtin_amdgcn_s_wait_asynccnt(0);
#else
  asm volatile("s_wait_asynccnt 0x0" ::: "memory");
#endif
#endif
}

// =====================================================================
// WMMA GEMM: C[M x 128] = act(A[M x Kd] * B[Kd x 128] + bias)
// One wave computes a 16x16 f32 tile with V_WMMA_F32_16X16X4_F32.
// 8 waves / 256-thread block -> 32 rows x 64 cols per block; grid.y = 2.
// B panel staged in LDS via GLOBAL_LOAD_ASYNC_TO_LDS_B128 (K-chunks of 64).
// Epilogue additionally emits deterministic per-block column sum / sum^2
// partials for the following training-mode BatchNorm (fused stats pass).
// =====================================================================
__global__ __launch_bounds__(256) void gemm_wmma_f32(
    const float* __restrict__ A, int lda,
    const float* __restrict__ B, int ldb,
    const float* __restrict__ bias,
    float* __restrict__ C, int ldc,
    int M, int Kd, int act,
    float* __restrict__ psum, float* __restrict__ psumsq)
{
  __shared__ float Bs[K_CHUNK * 64];     // 16 KB B panel chunk
  __shared__ float redS[256];            // cross-wave column-sum reduce
  __shared__ float redQ[256];

  const int wave = threadIdx.x >> 5;
  const int lane = threadIdx.x & 31;
  const int lo   = lane & 15;
  const int hi   = lane >> 4;                       // 0 or 1
  const int tileM   = blockIdx.x * 32 + (wave & 1) * 16;
  const int nbase   = blockIdx.y * 64;
  const int col_loc = (wave >> 1) * 16 + lo;        // 0..63 within block

  // A fragment pointer: lanes 0-15 hold K=+0,+1; lanes 16-31 hold K=+2,+3
  const float* ap = A + (size_t)(tileM + lo) * lda + 2 * hi;

  v8f acc = {0.f, 0.f, 0.f, 0.f, 0.f, 0.f, 0.f, 0.f};

  for (int k0 = 0; k0 < Kd; k0 += K_CHUNK) {
    // ---- cooperative load of B[k0:k0+64, nbase:nbase+64] into LDS ----
#pragma unroll
    for (int it = 0; it < 4; ++it) {
      int f    = it * 1024 + threadIdx.x * 4;       // float index in chunk
      int krow = f >> 6;
      int c    = f & 63;
      const float* srcp = B + (size_t)(k0 + krow) * ldb + nbase + c;
#if USE_ASYNC_LDS
      __builtin_amdgcn_global_load_async_to_lds_b128(
          (glb_v4i*)srcp, (lds_v4i*)&Bs[f], 0, 0);
#else
      *(float4*)&Bs[f] = *(const float4*)srcp;
#endif
    }
    wait_async_lds();
    __syncthreads();

    const float* apk = ap + k0;
#pragma unroll
    for (int kk = 0; kk < K_CHUNK; kk += 4) {
      v2f a = *(const v2f*)(apk + kk);              // contiguous, 8B aligned
      v2f b;
      b.x = Bs[(kk + 2 * hi) * 64 + col_loc];
      b.y = Bs[(kk + 2 * hi + 1) * 64 + col_loc];
      acc = __builtin_amdgcn_wmma_f32_16x16x4_f32(
          /*neg_a=*/false, a, /*neg_b=*/false, b,
          /*c_mod=*/(short)0, acc, /*reuse_a=*/false, /*reuse_b=*/false);
    }
    __syncthreads();                                // before Bs overwrite
  }

  // ---- epilogue: bias + activation + store + BN column partials ----
  const int col = nbase + col_loc;
  const float bv = bias ? bias[col] : 0.0f;
  float s = 0.f, q = 0.f;
#pragma unroll
  for (int r = 0; r < 8; ++r) {
    int row = tileM + r + 8 * hi;
    if (row < M) {
      float v = acc[r] + bv;
      if (act == 1) v = tanhf(v);
      C[(size_t)row * ldc + col] = v;
      s += v;
      q += v * v;
    }
  }
  redS[threadIdx.x] = s;
  redQ[threadIdx.x] = q;
  __syncthreads();
  if (threadIdx.x < 64) {
    int cb = threadIdx.x;                           // block-local column
    int w0 = (cb >> 4) * 2;                         // wave pair for this col
    int l  = cb & 15;
    float S = redS[w0 * 32 + l] + redS[w0 * 32 + l + 16]
            + redS[(w0 + 1) * 32 + l] + redS[(w0 + 1) * 32 + l + 16];
    float Q = redQ[w0 * 32 + l] + redQ[w0 * 32 + l + 16]
            + redQ[(w0 + 1) * 32 + l] + redQ[(w0 + 1) * 32 + l + 16];
    psum[(size_t)blockIdx.x * D_DIM + nbase + cb]   = S;
    psumsq[(size_t)blockIdx.x * D_DIM + nbase + cb] = Q;
  }
}

// ---------------- elementwise helpers ----------------
__global__ void vec_copy(const float* __restrict__ a, float* __restrict__ y, long long n4)
{
  long long t = (long long)blockIdx.x * blockDim.x + threadIdx.x;
  if (t < n4) ((float4*)y)[t] = ((const float4*)a)[t];
}

__global__ void vec_add(const float* __restrict__ a, const float* __restrict__ b,
                        float* __restrict__ y, long long n4)
{
  long long t = (long long)blockIdx.x * blockDim.x + threadIdx.x;
  if (t < n4) {
    float4 x = ((const float4*)a)[t], z = ((const float4*)b)[t];
    float4 o = {x.x + z.x, x.y + z.y, x.z + z.z, x.w + z.w};
    ((float4*)y)[t] = o;
  }
}

__global__ void vec_zero(float* __restrict__ y, long long n)
{
  long long t = (long long)blockIdx.x * blockDim.x + threadIdx.x;
  if (t < n) y[t] = 0.0f;
}

// ---------------- spmm scatter: pooled[dst] += x[src] ----------------
__global__ void edge_scatter(const float* __restrict__ x, int ldx,
                             const int* __restrict__ src, const int* __restrict__ dst,
                             float* __restrict__ pooled)
{
  long long t = (long long)blockIdx.x * blockDim.x + threadIdx.x;
  const int per = D_DIM / 4;                                  // 32 float4 per edge
  if (t >= (long long)E_EDGES * per) return;
  int e = (int)(t / per);
  int j = (int)(t % per) * 4;
  int s = src[e], d = dst[e];
  float4 v = *(const float4*)(x + (size_t)s * ldx + j);
  float* pp = pooled + (size_t)d * D_DIM + j;
  atomicAdd(pp + 0, v.x);
  atomicAdd(pp + 1, v.y);
  atomicAdd(pp + 2, v.z);
  atomicAdd(pp + 3, v.w);
}

// ---------------- BatchNorm (training mode, biased var) ----------------
// stats pass for the small [N,H] tensor only; big tensors fuse stats in GEMM
__global__ void bn_stats_partial(const float* __restrict__ x, int ld, int C, int Nrows,
                                 float* __restrict__ psum, float* __restrict__ psumsq)
{
  int c = threadIdx.x;
  if (c >= C) return;
  int r0 = blockIdx.x * BN_ROWS;
  int rend = r0 + BN_ROWS; if (rend > Nrows) rend = Nrows;
  float s = 0.f, s2 = 0.f;
  for (int r = r0; r < rend; ++r) {
    float v = x[(size_t)r * ld + c];
    s += v; s2 += v * v;
  }
  psum[(size_t)blockIdx.x * C + c]   = s;
  psumsq[(size_t)blockIdx.x * C + c] = s2;
}

__global__ void bn_finalize(const float* __restrict__ psum, const float* __restrict__ psumsq,
                            int nblk, int C, float invN,
                            float* __restrict__ mean, float* __restrict__ invstd)
{
  int c = blockIdx.x * blockDim.x + threadIdx.x;
  if (c >= C) return;
  float s = 0.f, s2 = 0.f;
  for (int i = 0; i < nblk; ++i) {          // fixed order -> deterministic
    s  += psum[(size_t)i * C + c];
    s2 += psumsq[(size_t)i * C + c];
  }
  float m = s * invN;
  float v = s2 * invN - m * m;
  mean[c]   = m;
  invstd[c] = rsqrtf(v + BN_EPS);
}

__global__ void bn_apply(const float* __restrict__ x, int ldi,
                         float* __restrict__ y, int ldo,
                         const float* __restrict__ mean, const float* __restrict__ invstd,
                         const float* __restrict__ g, const float* __restrict__ b,
                         int C, int Nrows, int relu)
{
  long long t = (long long)blockIdx.x * blockDim.x + threadIdx.x;
  if (t >= (long long)Nrows * C) return;
  int r = (int)(t / C), c = (int)(t % C);
  float v = (x[(size_t)r * ldi + c] - mean[c]) * invstd[c] * g[c] + b[c];
  if (relu) v = fmaxf(v, 0.0f);
  y[(size_t)r * ldo + c] = v;
}

// ---------------- attention head projection: [N,128] x [128,5] ----------------
__global__ void attn_head_gemm(const float* __restrict__ a, const float* __restrict__ W2,
                               const float* __restrict__ b2, float* __restrict__ ah)
{
  int t = blockIdx.x * blockDim.x + threadIdx.x;
  int n = t >> 3, h = t & 7;
  if (n >= N_NODES || h >= H_HEADS) return;
  const float* ar = a + (size_t)n * D_DIM;
  float s = b2[h];
  for (int d = 0; d < D_DIM; ++d) s += ar[d] * W2[d * H_HEADS + h];
  ah[(size_t)n * H_HEADS + h] = s;
}

__global__ void head_softmax(const float* __restrict__ a, float* __restrict__ alpha)
{
  int n = blockIdx.x * blockDim.x + threadIdx.x;
  if (n >= N_NODES) return;
  float v[H_HEADS], m = -3.0e38f;
#pragma unroll
  for (int h = 0; h < H_HEADS; ++h) { v[h] = a[(size_t)n * H_HEADS + h]; m = fmaxf(m, v[h]); }
  float s = 0.f;
#pragma unroll
  for (int h = 0; h < H_HEADS; ++h) { v[h] = __expf(v[h] - m); s += v[h]; }
  float inv = 1.0f / s;
#pragma unroll
  for (int h = 0; h < H_HEADS; ++h) alpha[(size_t)n * H_HEADS + h] = v[h] * inv;
}

// graph_emb[g,h,d] += alpha[n,h] * emb[n,d]
__global__ void pool_scatter(const float* __restrict__ emb, const float* __restrict__ alpha,
                             const int* __restrict__ seg, float* __restrict__ graph)
{
  long long t = (long long)blockIdx.x * blockDim.x + threadIdx.x;
  if (t >= (long long)N_NODES * D_DIM) return;
  int n = (int)(t / D_DIM), d = (int)(t % D_DIM);
  float e = emb[(size_t)n * D_DIM + d];
  float* gp = graph + (size_t)seg[n] * (H_HEADS * D_DIM) + d;
#pragma unroll
  for (int h = 0; h < H_HEADS; ++h)
    atomicAdd(gp + h * D_DIM, alpha[(size_t)n * H_HEADS + h] * e);
}

__global__ void relu_out(const float* __restrict__ graph, float* __restrict__ out, long long n)
{
  long long t = (long long)blockIdx.x * blockDim.x + threadIdx.x;
  if (t < n) out[t] = fmaxf(graph[t], 0.0f);
}

// L1 reg over 5 weight tensors, deterministic single-block reduction
__global__ __launch_bounds__(256) void l1_reg(
    const float* w0, int n0, const float* w1, int n1, const float* w2, int n2,
    const float* w3, int n3, const float* w4, int n4, float* __restrict__ out)
{
  __shared__ float sm[256];
  float s = 0.f;
  for (int i = threadIdx.x; i < n0; i += 256) s += fabsf(w0[i]);
  for (int i = threadIdx.x; i < n1; i += 256) s += fabsf(w1[i]);
  for (int i = threadIdx.x; i < n2; i += 256) s += fabsf(w2[i]);
  for (int i = threadIdx.x; i < n3; i += 256) s += fabsf(w3[i]);
  for (int i = threadIdx.x; i < n4; i += 256) s += fabsf(w4[i]);
  sm[threadIdx.x] = s;
  __syncthreads();
  for (int off = 128; off > 0; off >>= 1) {
    if ((int)threadIdx.x < off) sm[threadIdx.x] += sm[threadIdx.x + off];
    __syncthreads();
  }
  if (threadIdx.x == 0) out[0] = sm[0] / (float)G_GRAPHS;
}

// =====================================================================
extern "C" void kernel_launch(void* const* d_in, const int* in_sizes, int n_in,
                              void* d_out, int out_size, void* d_ws, size_t ws_size,
                              hipStream_t stream)
{
  (void)in_sizes; (void)n_in; (void)out_size; (void)ws_size;

  const float* node_feat = (const float*)d_in[0];
  const int*   src       = (const int*)d_in[1];
  const int*   dst       = (const int*)d_in[2];
  const int*   seg       = (const int*)d_in[3];
  const float* node_W    = (const float*)d_in[4];
  const float* node_b    = (const float*)d_in[5];
  const float* bn1_g     = (const float*)d_in[6];
  const float* bn1_b     = (const float*)d_in[7];
  const float* conv_W    = (const float*)d_in[8];
  const float* conv_b    = (const float*)d_in[9];
  const float* bn2_g     = (const float*)d_in[10];
  const float* bn2_b     = (const float*)d_in[11];
  const float* k_weight  = (const float*)d_in[12];
  const float* bn3_g     = (const float*)d_in[13];
  const float* bn3_b     = (const float*)d_in[14];
  const float* attn_W1   = (const float*)d_in[15];
  const float* attn_b1   = (const float*)d_in[16];
  const float* bna_g     = (const float*)d_in[17];
  const float* bna_b     = (const float*)d_in[18];
  const float* attn_W2   = (const float*)d_in[19];
  const float* attn_b2   = (const float*)d_in[20];
  const float* bnb_g     = (const float*)d_in[21];
  const float* bnb_b     = (const float*)d_in[22];

  float* out = (float*)d_out;
  float* ws  = (float*)d_ws;

  // ---------------- workspace layout (floats) ----------------
  const size_t ND = (size_t)N_NODES * D_DIM;              // 12.8M
  float* inp    = ws;                                     // [N,D]
  float* cur    = ws + 1 * ND;                            // [N,D]
  float* bibuf  = ws + 2 * ND;                            // [N,D] block_input / attn buf
  float* pooled = ws + 3 * ND;                            // [N,D]
  float* tmp    = ws + 4 * ND;                            // [N,D] gemm output pre-BN
  float* outs   = ws + 5 * ND;                            // [N,3D]
  float* small  = ws + 8 * ND;
  float* psum   = small;                    small += (size_t)GEMM_ROW_BLOCKS * D_DIM;
  float* psumsq = small;                    small += (size_t)GEMM_ROW_BLOCKS * D_DIM;
  float* meanb  = small;                    small += D_DIM;
  float* invstd = small;                    small += D_DIM;
  float* ah     = small;                    small += (size_t)N_NODES * H_HEADS;
  float* ahn    = small;                    small += (size_t)N_NODES * H_HEADS;
  float* alpha  = small;                    small += (size_t)N_NODES * H_HEADS;
  float* graph  = small;                    // [G,H,D] = 640000

  const long long ND4 = (long long)ND / 4;
  const float invN = 1.0f / (float)N_NODES;
  const int nblk_h = (N_NODES + BN_ROWS - 1) / BN_ROWS;   // 196 (head BN only)

  auto gemm = [&](const float* A, int lda, const float* B, int ldb, const float* bias,
                  float* C, int ldc, int Kd, int act) {
    dim3 grid(GEMM_ROW_BLOCKS, D_DIM / 64);
    gemm_wmma_f32<<<grid, 256, 0, stream>>>(A, lda, B, ldb, bias, C, ldc,
                                            N_NODES, Kd, act, psum, psumsq);
  };
  // BN after a fused-stats GEMM: finalize partials (fixed order) + apply
  auto bn = [&](const float* x, int ldi, float* y, int ldo,
                const float* g, const float* b, int relu) {
    bn_finalize<<<1, D_DIM, 0, stream>>>(psum, psumsq, GEMM_ROW_BLOCKS, D_DIM,
                                         invN, meanb, invstd);
    long long tot = (long long)N_NODES * D_DIM;
    bn_apply<<<(unsigned)((tot + 255) / 256), 256, 0, stream>>>(
        x, ldi, y, ldo, meanb, invstd, g, b, D_DIM, N_NODES, relu);
  };

  // ---- node embedding: relu(bn(node_feat @ node_W + node_b)) ----
  gemm(node_feat, F_INF, node_W, D_DIM, node_b, tmp, D_DIM, F_INF, 0);
  bn(tmp, D_DIM, inp, D_DIM, bn1_g, bn1_b, /*relu=*/1);

  // ---- 3 DAGCN blocks ----
  const long long escat_tot = (long long)E_EDGES * (D_DIM / 4);
  for (int blk = 0; blk < NUM_BLOCKS; ++blk) {
    const float* bi;
    if (blk == 0) {
      bi = inp;                                           // cur == inp at block 0
    } else {
      vec_add<<<(unsigned)((ND4 + 255) / 256), 256, 0, stream>>>(cur, inp, bibuf, ND4);
      bi = bibuf;
    }
    const float* x = bi; int ldx = D_DIM;
    for (int k = 0; k < K_HOPS; ++k) {
      // pooled = spmm(x) + block_input
      vec_copy<<<(unsigned)((ND4 + 255) / 256), 256, 0, stream>>>(bi, pooled, ND4);
      edge_scatter<<<(unsigned)((escat_tot + 255) / 256), 256, 0, stream>>>(
          x, ldx, src, dst, pooled);
      // input_x = bn(pooled @ conv_W[k] + conv_b[k])  -> outs[:, k*D:(k+1)*D]
      gemm(pooled, D_DIM, conv_W + (size_t)k * D_DIM * D_DIM, D_DIM,
           conv_b + (size_t)k * D_DIM, tmp, D_DIM, D_DIM, 0);
      bn(tmp, D_DIM, outs + (size_t)k * D_DIM, K_HOPS * D_DIM,
         bn2_g + (size_t)k * D_DIM, bn2_b + (size_t)k * D_DIM, /*relu=*/0);
      x = outs + (size_t)k * D_DIM; ldx = K_HOPS * D_DIM;
    }
    // cur = relu(bn(outs @ k_weight))
    gemm(outs, K_HOPS * D_DIM, k_weight, D_DIM, nullptr, tmp, D_DIM, K_HOPS * D_DIM, 0);
    bn(tmp, D_DIM, cur, D_DIM, bn3_g, bn3_b, /*relu=*/1);
  }

  // ---- attention pooling ----
  // a = bn(tanh(cur @ attn_W1 + attn_b1))
  gemm(cur, D_DIM, attn_W1, D_DIM, attn_b1, tmp, D_DIM, D_DIM, /*act=tanh*/1);
  bn(tmp, D_DIM, bibuf, D_DIM, bna_g, bna_b, /*relu=*/0);
  // a = bn(a @ attn_W2 + attn_b2) ; softmax over heads
  {
    long long tot = (long long)N_NODES * 8;
    attn_head_gemm<<<(unsigned)((tot + 255) / 256), 256, 0, stream>>>(bibuf, attn_W2, attn_b2, ah);
    bn_stats_partial<<<nblk_h, 32, 0, stream>>>(ah, H_HEADS, H_HEADS, N_NODES, psum, psumsq);
    bn_finalize<<<1, 32, 0, stream>>>(psum, psumsq, nblk_h, H_HEADS, invN, meanb, invstd);
    long long tot2 = (long long)N_NODES * H_HEADS;
    bn_apply<<<(unsigned)((tot2 + 255) / 256), 256, 0, stream>>>(
        ah, H_HEADS, ahn, H_HEADS, meanb, invstd, bnb_g, bnb_b, H_HEADS, N_NODES, 0);
    head_softmax<<<(N_NODES + 255) / 256, 256, 0, stream>>>(ahn, alpha);
  }
  // graph_emb = segment_sum(alpha[:, :, None] * node_emb[:, None, :])
  {
    const long long gsize = (long long)G_GRAPHS * H_HEADS * D_DIM;
    vec_zero<<<(unsigned)((gsize + 255) / 256), 256, 0, stream>>>(graph, gsize);
    long long tot = (long long)N_NODES * D_DIM;
    pool_scatter<<<(unsigned)((tot + 255) / 256), 256, 0, stream>>>(cur, alpha, seg, graph);
    relu_out<<<(unsigned)((gsize + 255) / 256), 256, 0, stream>>>(graph, out, gsize);
  }
  // ---- L1 regularizer / G ----
  l1_reg<<<1, 256, 0, stream>>>(node_W, F_INF * D_DIM,
                                conv_W, K_HOPS * D_DIM * D_DIM,
                                k_weight, K_HOPS * D_DIM * D_DIM,
                                attn_W1, D_DIM * D_DIM,
                                attn_W2, D_DIM * H_HEADS,
                                out + (size_t)G_GRAPHS * H_HEADS * D_DIM);
}